// SuperpixelGAT_15762529976323
// MI455X (gfx1250) — compile-verified
//
#include <hip/hip_runtime.h>
#include <math.h>

typedef float v2f __attribute__((ext_vector_type(2)));
typedef float v8f __attribute__((ext_vector_type(8)));

#define HD 30
#define NG 256
#define OUTS 10

static __device__ __forceinline__ void atomicMaxF(float* addr, float val) {
  if (!(val < 0.0f)) {
    atomicMax((int*)addr, __float_as_int(val));
  } else {
    atomicMin((unsigned int*)addr, __float_as_uint(val));
  }
}

__global__ void gat_fill(float* p, float v, long n) {
  long i = (long)blockIdx.x * blockDim.x + threadIdx.x;
  if (i < n) p[i] = v;
}

// Padded-head column map: col c in [0,64): c<30 -> head0 feat c; 32<=c<62 -> head1 feat c-32; else pad.
// Pack W[l] into Wp[l] ([128][64], zero padded) and bp[l] ([64]).
// For layers 1/2 the K dimension is also the padded-64 layout (input rows are padded).
__global__ void gat_pack(const float* W1, const float* b1,
                         const float* W2, const float* b2,
                         const float* W3, const float* b3,
                         float* Wp, float* bp) {
  int t = blockIdx.x * blockDim.x + threadIdx.x;
  if (t >= 3 * 128 * 64) return;
  int l = t / (128 * 64);
  int r = t % (128 * 64);
  int k = r / 64, c = r % 64;
  int kdim = (l == 0) ? 128 : 60;
  const float* W = (l == 0) ? W1 : ((l == 1) ? W2 : W3);
  const float* b = (l == 0) ? b1 : ((l == 1) ? b2 : b3);

  // map padded k index -> dense k index (layer 0: identity over 128)
  int kk;
  if (l == 0) kk = (k < 128) ? k : -1;
  else        kk = (k < 30) ? k : ((k >= 32 && k < 62) ? k - 2 : -1);
  // map padded col -> dense concat col
  int cc = (c < 30) ? c : ((c >= 32 && c < 62) ? c - 2 : -1);

  float v = 0.0f;
  if (kk >= 0 && cc >= 0) {
    int h = cc / HD, j = cc % HD;
    v = W[(h * kdim + kk) * HD + j];
  }
  Wp[t] = v;
  if (k == 0) {
    float bv = 0.0f;
    if (cc >= 0) bv = b[(cc / HD) * HD + (cc % HD)];
    bp[l * 64 + c] = bv;
  }
}

// One wave per 16-node tile; fp32 WMMA 16x16x4, 4 output col-tiles (64 cols).
// Weights staged in LDS, pair-interleaved so each B fragment is one ds_load_b64.
// feat layout: [N][64], cols 0-29 head0, 32-61 head1, pads zero.
__global__ void gat_feat_wmma(const float* __restrict__ xin, int ldx, int K,
                              const float* __restrict__ Wp,
                              const float* __restrict__ bp,
                              float* __restrict__ feat, int N) {
  __shared__ float lWp[8192];   // [pair p][col c][k&1] = Wp[k][c], p = k>>1
  __shared__ float lbp[64];
  int tid = threadIdx.x;
  for (int i = tid; i < K * 64; i += blockDim.x) {
    int k = i >> 6, c = i & 63;
    lWp[(k >> 1) * 128 + c * 2 + (k & 1)] = Wp[i];
  }
  if (tid < 64) lbp[tid] = bp[tid];
  __syncthreads();

  int wave = (blockIdx.x * blockDim.x + tid) >> 5;
  int lane = tid & 31;
  int row0 = wave * 16;
  if (row0 >= N) return;            // uniform per wave: EXEC stays all-1 for WMMA
  int lo = lane & 15, hi = lane >> 4;
  int row = row0 + lo;
  bool rowok = (row < N);
  const float* arow = xin + (long)(rowok ? row : row0) * ldx;
  v8f c0 = {}, c1 = {}, c2 = {}, c3 = {};
  for (int k = 0; k < K; k += 4) {
    v2f a = *(const v2f*)(arow + k + 2 * hi);         // global_load_b64
    if (!rowok) { a.x = 0.0f; a.y = 0.0f; }
    const float* bbase = lWp + (k / 2 + hi) * 128;
    v2f b0 = *(const v2f*)(bbase + 2 * lo);           // ds_load_b64
    v2f b1 = *(const v2f*)(bbase + 2 * (16 + lo));
    v2f b2 = *(const v2f*)(bbase + 2 * (32 + lo));
    v2f b3 = *(const v2f*)(bbase + 2 * (48 + lo));
    c0 = __builtin_amdgcn_wmma_f32_16x16x4_f32(false, a, false, b0, (short)0, c0, false, false);
    c1 = __builtin_amdgcn_wmma_f32_16x16x4_f32(false, a, false, b1, (short)0, c1, false, false);
    c2 = __builtin_amdgcn_wmma_f32_16x16x4_f32(false, a, false, b2, (short)0, c2, false, false);
    c3 = __builtin_amdgcn_wmma_f32_16x16x4_f32(false, a, false, b3, (short)0, c3, false, false);
  }
  // D layout: VGPR r holds (M = r + 8*hi, N = lo)
  #pragma unroll
  for (int r = 0; r < 8; ++r) {
    int m = row0 + r + 8 * hi;
    if (m < N) {
      float* frow = feat + (long)m * 64;
      frow[lo]      = c0[r] + lbp[lo];
      frow[16 + lo] = c1[r] + lbp[16 + lo];
      frow[32 + lo] = c2[r] + lbp[32 + lo];
      frow[48 + lo] = c3[r] + lbp[48 + lo];
    }
  }
}

// a1[h][n] = feat_h[n] . s1w[h] + s1b[h]; a2 likewise (feat head h at col h*32)
__global__ void gat_a12(const float* __restrict__ feat,
                        const float* __restrict__ s1w, const float* __restrict__ s1b,
                        const float* __restrict__ s2w, const float* __restrict__ s2b,
                        float* __restrict__ a1, float* __restrict__ a2, int N) {
  int n = blockIdx.x * blockDim.x + threadIdx.x;
  if (n >= N) return;
  const float* f = feat + (long)n * 64;
  #pragma unroll
  for (int h = 0; h < 2; ++h) {
    float d1 = s1b[h], d2 = s2b[h];
    #pragma unroll
    for (int j = 0; j < HD; ++j) {
      float v = f[h * 32 + j];
      d1 = fmaf(v, s1w[h * HD + j], d1);
      d2 = fmaf(v, s2w[h * HD + j], d2);
    }
    a1[h * N + n] = d1;
    a2[h * N + n] = d2;
  }
}

__global__ void gat_edge_max(const int* __restrict__ src, const int* __restrict__ dst,
                             const float* __restrict__ a1, const float* __restrict__ a2,
                             float* __restrict__ ebuf, float* __restrict__ m,
                             int E, int N) {
  int t = blockIdx.x * blockDim.x + threadIdx.x;
  if (t >= 2 * E) return;
  int h = (t < E) ? 0 : 1;
  int e = t - h * E;
  int s = src[e], d = dst[e];
  float v = a1[h * N + s] + a2[h * N + d];
  v = (v > 0.0f) ? v : 0.01f * v;   // leaky_relu
  ebuf[t] = v;
  atomicMaxF(&m[h * N + s], v);
}

__global__ void gat_edge_exp(const int* __restrict__ src,
                             const float* __restrict__ m, float* __restrict__ ebuf,
                             float* __restrict__ den, int E, int N) {
  int t = blockIdx.x * blockDim.x + threadIdx.x;
  if (t >= 2 * E) return;
  int h = (t < E) ? 0 : 1;
  int e = t - h * E;
  int s = src[e];
  float ev = __expf(ebuf[t] - m[h * N + s]);
  ebuf[t] = ev;
  atomicAdd(&den[h * N + s], ev);
}

__global__ void gat_edge_agg(const int* __restrict__ src, const int* __restrict__ dst,
                             const float* __restrict__ ebuf, const float* __restrict__ den,
                             const float* __restrict__ feat, float* __restrict__ agg,
                             int E, int N) {
  int t = blockIdx.x * blockDim.x + threadIdx.x;
  if (t >= 2 * E) return;
  int h = (t < E) ? 0 : 1;
  int e = t - h * E;
  int s = src[e], d = dst[e];
  float attn = ebuf[t] / den[h * N + s];
  const float* fd = feat + (long)d * 64 + h * 32;   // 16B aligned
  float* os = agg + (long)s * 64 + h * 32;
  const float4* fd4 = (const float4*)fd;
  #pragma unroll
  for (int j4 = 0; j4 < 7; ++j4) {                  // 28 floats via global_load_b128
    float4 q = fd4[j4];
    atomicAdd(&os[j4 * 4 + 0], attn * q.x);
    atomicAdd(&os[j4 * 4 + 1], attn * q.y);
    atomicAdd(&os[j4 * 4 + 2], attn * q.z);
    atomicAdd(&os[j4 * 4 + 3], attn * q.w);
  }
  v2f tail = *(const v2f*)(fd + 28);
  atomicAdd(&os[28], attn * tail.x);
  atomicAdd(&os[29], attn * tail.y);
}

__global__ void gat_relu_out(const float* __restrict__ agg, float* __restrict__ xout, long n64) {
  long i = (long)blockIdx.x * blockDim.x + threadIdx.x;
  if (i >= n64) return;
  int c = (int)(i & 63);
  bool valid = (c < 30) || (c >= 32 && c < 62);
  float v = valid ? agg[i] : 0.0f;
  xout[i] = (v > 0.0f) ? v : 0.0f;
}

__global__ void gat_pool(const int* __restrict__ batch,
                         const float* __restrict__ x1, const float* __restrict__ x2,
                         const float* __restrict__ x3,
                         float* __restrict__ pooled, float* __restrict__ counts, int N) {
  int n = blockIdx.x * blockDim.x + threadIdx.x;
  if (n >= N) return;
  int g = batch[n];
  float* pg = pooled + (long)g * 180;
  const float* rows[3] = { x1 + (long)n * 64, x2 + (long)n * 64, x3 + (long)n * 64 };
  #pragma unroll
  for (int l = 0; l < 3; ++l) {
    const float* r = rows[l];
    for (int c = 0; c < 64; ++c) {
      bool valid = (c < 30) || (c >= 32 && c < 62);
      if (valid) {
        int f = (c < 30) ? c : c - 2;               // dense concat index
        atomicAdd(&pg[l * 60 + f], r[c]);
      }
    }
  }
  atomicAdd(&counts[g], 1.0f);
}

__global__ void gat_head(const float* __restrict__ pooled, const float* __restrict__ counts,
                         const float* __restrict__ Wf, const float* __restrict__ bf,
                         float* __restrict__ out) {
  int g = threadIdx.x;
  if (g >= NG) return;
  float cnt = counts[g];
  cnt = (cnt > 1.0f) ? cnt : 1.0f;
  float inv = 1.0f / cnt;
  float logit[OUTS];
  #pragma unroll
  for (int o = 0; o < OUTS; ++o) logit[o] = bf[o];
  const float* pg = pooled + (long)g * 180;
  for (int f = 0; f < 180; ++f) {
    float p = pg[f] * inv;
    #pragma unroll
    for (int o = 0; o < OUTS; ++o) logit[o] = fmaf(p, Wf[f * OUTS + o], logit[o]);
  }
  float mx = logit[0];
  #pragma unroll
  for (int o = 1; o < OUTS; ++o) mx = fmaxf(mx, logit[o]);
  float sum = 0.0f;
  #pragma unroll
  for (int o = 0; o < OUTS; ++o) { logit[o] = __expf(logit[o] - mx); sum += logit[o]; }
  float r = 1.0f / sum;
  #pragma unroll
  for (int o = 0; o < OUTS; ++o) out[g * OUTS + o] = logit[o] * r;
}

static inline int gridFor(long n, int b) { return (int)((n + b - 1) / b); }

extern "C" void kernel_launch(void* const* d_in, const int* in_sizes, int n_in,
                              void* d_out, int out_size, void* d_ws, size_t ws_size,
                              hipStream_t stream) {
  const float* x     = (const float*)d_in[0];
  const int*   idx   = (const int*)d_in[1];
  const int*   batch = (const int*)d_in[2];
  const int N = in_sizes[0] / 128;
  const int E = in_sizes[1] / 2;
  const int* src = idx;
  const int* dst = idx + E;

  const float* W[3];  const float* b[3];
  const float* s1w[3]; const float* s1b[3];
  const float* s2w[3]; const float* s2b[3];
  for (int l = 0; l < 3; ++l) {
    W[l]   = (const float*)d_in[3 + 6 * l + 0];
    b[l]   = (const float*)d_in[3 + 6 * l + 1];
    s1w[l] = (const float*)d_in[3 + 6 * l + 2];
    s1b[l] = (const float*)d_in[3 + 6 * l + 3];
    s2w[l] = (const float*)d_in[3 + 6 * l + 4];
    s2b[l] = (const float*)d_in[3 + 6 * l + 5];
  }
  const float* Wf = (const float*)d_in[21];
  const float* bf = (const float*)d_in[22];

  // workspace layout (floats)
  float* base   = (float*)d_ws;
  float* Wp     = base;                      // 3*128*64
  float* bp     = Wp + 3 * 128 * 64;         // 3*64
  float* feat   = bp + 3 * 64;               // N*64
  float* a1     = feat + (long)N * 64;       // 2N
  float* a2     = a1 + 2L * N;               // 2N
  float* mval   = a2 + 2L * N;               // 2N
  float* den    = mval + 2L * N;             // 2N
  float* ebuf   = den + 2L * N;              // 2E
  float* agg    = ebuf + 2L * E;             // N*64
  float* xo0    = agg + (long)N * 64;        // N*64
  float* xo1    = xo0 + (long)N * 64;        // N*64
  float* xo2    = xo1 + (long)N * 64;        // N*64
  float* pooled = xo2 + (long)N * 64;        // 256*180
  float* counts = pooled + NG * 180;         // 256
  float* xo[3] = { xo0, xo1, xo2 };

  const int B = 256;
  const float NEG_INF = -__builtin_huge_valf();

  gat_pack<<<gridFor(3 * 128 * 64, B), B, 0, stream>>>(W[0], b[0], W[1], b[1], W[2], b[2], Wp, bp);

  for (int l = 0; l < 3; ++l) {
    const float* xin = (l == 0) ? x : xo[l - 1];
    int ldx = (l == 0) ? 128 : 64;
    int K   = (l == 0) ? 128 : 64;

    gat_fill<<<gridFor(2L * N, B), B, 0, stream>>>(mval, NEG_INF, 2L * N);
    gat_fill<<<gridFor(2L * N, B), B, 0, stream>>>(den, 0.0f, 2L * N);
    gat_fill<<<gridFor((long)N * 64, B), B, 0, stream>>>(agg, 0.0f, (long)N * 64);

    int tiles = (N + 15) / 16;                 // one wave (32 lanes) per 16-row tile
    gat_feat_wmma<<<gridFor((long)tiles * 32, B), B, 0, stream>>>(
        xin, ldx, K, Wp + (long)l * 128 * 64, bp + l * 64, feat, N);

    gat_a12<<<gridFor(N, B), B, 0, stream>>>(feat, s1w[l], s1b[l], s2w[l], s2b[l], a1, a2, N);

    gat_edge_max<<<gridFor(2L * E, B), B, 0, stream>>>(src, dst, a1, a2, ebuf, mval, E, N);
    gat_edge_exp<<<gridFor(2L * E, B), B, 0, stream>>>(src, mval, ebuf, den, E, N);
    gat_edge_agg<<<gridFor(2L * E, B), B, 0, stream>>>(src, dst, ebuf, den, feat, agg, E, N);

    gat_relu_out<<<gridFor((long)N * 64, B), B, 0, stream>>>(agg, xo[l], (long)N * 64);
  }

  gat_fill<<<gridFor((long)NG * 180 + NG, B), B, 0, stream>>>(pooled, 0.0f, (long)NG * 180 + NG);
  gat_pool<<<gridFor(N, B), B, 0, stream>>>(batch, xo0, xo1, xo2, pooled, counts, N);
  gat_head<<<1, NG, 0, stream>>>(pooled, counts, Wf, bf, (float*)d_out);
}